// GCN_En_49323404427441
// MI455X (gfx1250) — compile-verified
//
#include <hip/hip_runtime.h>
#include <hip/hip_bf16.h>

typedef __attribute__((ext_vector_type(2))) float v2f;
typedef __attribute__((ext_vector_type(8))) float v8f;

#define NNODES 50000
#define FDIM   128
#define HDIM   128
#define NEDGES 600000

// LDS strides padded to reduce bank conflicts (64 banks x 4B)
#define WSTRIDE 132
#define ASTRIDE 130

// ---------------- init: zero accumulator (d_out) and deg=1 (self loop) -------
__global__ __launch_bounds__(256) void gcn_init_kernel(float* __restrict__ out,
                                                       unsigned* __restrict__ deg) {
    int i = blockIdx.x * blockDim.x + threadIdx.x;
    if (i < NNODES * HDIM) out[i] = 0.0f;
    if (i < NNODES)        deg[i] = 1u;
}

// ---------------- degree: count incoming edges -------------------------------
__global__ __launch_bounds__(256) void gcn_deg_kernel(const long long* __restrict__ dst,
                                                      unsigned* __restrict__ deg) {
    int e = blockIdx.x * blockDim.x + threadIdx.x;
    if (e < NEDGES) atomicAdd(&deg[(int)dst[e]], 1u);
}

// ---------------- dinv = rsqrt(deg) ------------------------------------------
__global__ __launch_bounds__(256) void gcn_dinv_kernel(const unsigned* __restrict__ deg,
                                                       float* __restrict__ dinv) {
    int i = blockIdx.x * blockDim.x + threadIdx.x;
    if (i < NNODES) dinv[i] = rsqrtf((float)deg[i]);
}

// ---------------- WMMA fp32 GEMM: y = (x @ W) * dinv[row] --------------------
// One block = 16 rows of x. 8 waves, wave w owns columns [16w, 16w+16).
// Full W (128x128) + one 16x128 x-tile staged in LDS.
__global__ __launch_bounds__(256) void gcn_gemm_kernel(const float* __restrict__ x,
                                                       const float* __restrict__ W,
                                                       const float* __restrict__ dinv,
                                                       float* __restrict__ y) {
    __shared__ float lW[FDIM * WSTRIDE];   // 128 x 132 floats (~67.6 KB)
    __shared__ float lA[16 * ASTRIDE];     // 16 x 130 floats  (~8.3 KB)

    const int tid = threadIdx.x;
    const int m0  = blockIdx.x * 16;       // 50000 / 16 = 3125 blocks exactly

    // Stage W (row-major, padded) -- coalesced global reads
    for (int i = tid; i < FDIM * HDIM; i += 256) {
        int r = i >> 7, c = i & 127;
        lW[r * WSTRIDE + c] = W[i];
    }
    // Stage 16x128 tile of x
    for (int i = tid; i < 16 * FDIM; i += 256) {
        int r = i >> 7, c = i & 127;
        lA[r * ASTRIDE + c] = x[(size_t)(m0 + r) * FDIM + c];
    }
    __syncthreads();

    const int wave = tid >> 5;             // 0..7 -> n-tile
    const int lane = tid & 31;
    const int n0   = wave * 16;
    const int half = (lane >= 16) ? 2 : 0; // K sub-offset per ISA fp32 layout
    const int l15  = lane & 15;

    v8f acc = {};
    for (int k = 0; k < FDIM; k += 4) {
        v2f a, b;
        // A 16x4: lanes 0-15 hold K=k,k+1 of row M=lane; lanes 16-31 hold K=k+2,k+3
        a[0] = lA[l15 * ASTRIDE + k + half];
        a[1] = lA[l15 * ASTRIDE + k + half + 1];
        // B 4x16: lanes 0-15 hold rows K=k,k+1 at N=lane; lanes 16-31 rows K=k+2,k+3
        b[0] = lW[(k + half)     * WSTRIDE + n0 + l15];
        b[1] = lW[(k + half + 1) * WSTRIDE + n0 + l15];
        acc = __builtin_amdgcn_wmma_f32_16x16x4_f32(
            /*neg_a=*/false, a, /*neg_b=*/false, b,
            /*c_mod=*/(short)0, acc, /*reuse_a=*/false, /*reuse_b=*/false);
    }

    // C/D layout: lanes 0-15 -> M = i, lanes 16-31 -> M = 8+i (i = VGPR index)
    const int mbase = (lane < 16) ? 0 : 8;
    const int nout  = n0 + l15;
    #pragma unroll
    for (int i = 0; i < 8; ++i) {
        int m = m0 + mbase + i;
        y[(size_t)m * HDIM + nout] = acc[i] * dinv[m];
    }
}

// ---------------- scatter: acc[dst] += y[src] (wave per edge, float4/lane) ---
__global__ __launch_bounds__(256) void gcn_scatter_kernel(const long long* __restrict__ src,
                                                          const long long* __restrict__ dst,
                                                          const float* __restrict__ y,
                                                          float* __restrict__ acc) {
    int e    = (blockIdx.x * blockDim.x + threadIdx.x) >> 5;
    int lane = threadIdx.x & 31;
    if (e >= NEDGES) return;
    int s = (int)src[e];
    int t = (int)dst[e];
    const float4 v = ((const float4*)(y + (size_t)s * HDIM))[lane];
    float* o = acc + (size_t)t * HDIM + lane * 4;
    unsafeAtomicAdd(o + 0, v.x);   // -> global_atomic_add_f32 (L2-resident)
    unsafeAtomicAdd(o + 1, v.y);
    unsafeAtomicAdd(o + 2, v.z);
    unsafeAtomicAdd(o + 3, v.w);
}

// ---------------- finalize: out = relu((acc + y)*dinv[row] + b) --------------
__global__ __launch_bounds__(256) void gcn_final_kernel(float* __restrict__ out,
                                                        const float* __restrict__ y,
                                                        const float* __restrict__ dinv,
                                                        const float* __restrict__ b) {
    int i = blockIdx.x * blockDim.x + threadIdx.x;
    if (i < NNODES * HDIM) {
        int r = i >> 7, h = i & 127;
        float v = (out[i] + y[i]) * dinv[r] + b[h];
        out[i] = fmaxf(v, 0.0f);
    }
}

extern "C" void kernel_launch(void* const* d_in, const int* in_sizes, int n_in,
                              void* d_out, int out_size, void* d_ws, size_t ws_size,
                              hipStream_t stream) {
    const float*     x  = (const float*)d_in[0];           // [N, 128]
    const float*     W  = (const float*)d_in[1];           // [128, 128]
    const float*     b  = (const float*)d_in[2];           // [128]
    const long long* ei = (const long long*)d_in[3];       // [2, E] int64
    const long long* src = ei;
    const long long* dst = ei + NEDGES;
    float* out = (float*)d_out;                            // [N, 128]

    // Workspace layout: deg u32[N] | dinv f32[N] | y f32[N*128]
    unsigned* deg  = (unsigned*)d_ws;
    float*    dinv = (float*)((char*)d_ws + (size_t)NNODES * 4);
    float*    y    = (float*)((char*)d_ws + (size_t)NNODES * 8);   // 400000 B, 16B aligned

    const int elems = NNODES * HDIM;                       // 6.4M

    gcn_init_kernel<<<(elems + 255) / 256, 256, 0, stream>>>(out, deg);
    gcn_deg_kernel<<<(NEDGES + 255) / 256, 256, 0, stream>>>(dst, deg);
    gcn_dinv_kernel<<<(NNODES + 255) / 256, 256, 0, stream>>>(deg, dinv);
    gcn_gemm_kernel<<<NNODES / 16, 256, 0, stream>>>(x, W, dinv, y);
    gcn_scatter_kernel<<<(NEDGES * 32 + 255) / 256, 256, 0, stream>>>(src, dst, y, out);
    gcn_final_kernel<<<(elems + 255) / 256, 256, 0, stream>>>(out, y, dinv, b);
}